// WaveDecoder_42013370089835
// MI455X (gfx1250) — compile-verified
//
#include <hip/hip_runtime.h>
#include <stdint.h>

// ---------------------------------------------------------------------------
// Fused 3-level inverse Haar DWT (WaveDecoder), MI455X / gfx1250.
//
// Input  h  : (n, 192, 64, 64) f32   Output out : (n, 3, 512, 512) f32
// Channel packing:
//   [0:3)     ll0            (coarsest LL, 64x64)
//   [3:147)   level-0 detail: c*48 + b*16 + ph*4 + pw   (p=4 tiling of 256x256)
//   [147:183) level-1 detail: c*12 + b*4  + ph*2 + pw   (p=2 tiling of 128x128)
//   [183:192) level-2 detail: c*3  + b                  (64x64)
// Haar synthesis: out[2i+s,2j+t] = 0.5*(ll + (-1)^s lh + (-1)^t hl + (-1)^(s+t) hh)
//
// One thread -> one 4x4 output tile (all 16 descendants of one level-1
// position). Coarse row (ll0 + level-2 details) staged once per workgroup via
// gfx1250 async global->LDS DMA; every input element read from HBM exactly
// once. ~100 MB total traffic -> bandwidth-bound (~4.3 us @ 23.3 TB/s).
// c and n live in separate grid dims so no integer division is emitted.
// ---------------------------------------------------------------------------

typedef float v2f __attribute__((ext_vector_type(2)));
typedef float v4f __attribute__((ext_vector_type(4)));

namespace {
constexpr int kChanStride = 64 * 64;                   // elements per channel plane
constexpr int kInChannels = 192;
constexpr int kInNStride  = kInChannels * kChanStride; // elements per batch image
}

__global__ __launch_bounds__(256)
void haar_idwt3_fused(const float* __restrict__ h, float* __restrict__ out) {
  const int jt = threadIdx.x;          // 0..127  = level-1 col j1
  const int ty = threadIdx.y;          // 0..1
  const int c  = blockIdx.x;           // 0..2    = output channel
  const int by = blockIdx.y;           // 0..63   = coarse row i0
  const int n  = blockIdx.z;           // batch index
  const int it = (by << 1) | ty;       // 0..127  = level-1 row i1
  const int tid = (ty << 7) | jt;      // 0..255

  const float* __restrict__ hb = h + (size_t)n * kInNStride;

  // ---- Stage coarse coefficients (ll0 + level-2 details, row i0=by) into LDS
  //      via gfx1250 async global->LDS DMA. 4 arrays x 64 cols, 1 elem/thread.
  __shared__ float lds_coarse[4 * 64];
  {
    const int a  = tid >> 6;           // 0:ll0  1:lh2  2:hl2  3:hh2
    const int j0 = tid & 63;
    const int ch = (a == 0) ? c : (183 + c * 3 + (a - 1));
    const unsigned gbyte = (unsigned)((ch * kChanStride + by * 64 + j0) << 2);
    const unsigned lbyte =
        (unsigned)(uintptr_t)&lds_coarse[0] + (unsigned)(tid << 2);
    asm volatile("global_load_async_to_lds_b32 %0, %1, %2"
                 :
                 : "v"(lbyte), "v"(gbyte), "s"(hb)
                 : "memory");
  }
  asm volatile("s_wait_asynccnt 0" ::: "memory");
  __syncthreads();

  // ---- Level-2 synthesis: v1 = ll1(i1=it, j1=jt)
  const int j0t = jt >> 1;
  const float ll0 = lds_coarse[      j0t];
  const float lh2 = lds_coarse[ 64 + j0t];
  const float hl2 = lds_coarse[128 + j0t];
  const float hh2 = lds_coarse[192 + j0t];
  const float a0 = (it & 1) ? -1.f : 1.f;   // row parity sign
  const float b0 = (jt & 1) ? -1.f : 1.f;   // col parity sign
  const float v1 = 0.5f * (ll0 + a0 * lh2 + b0 * (hl2 + a0 * hh2));

  // ---- Level-1 details at (it, jt): bands are 4 channels apart
  float lh1, hl1, hh1;
  {
    const int ch1 = 147 + c * 12 + ((it >> 6) << 1) + (jt >> 6);
    const float* d1 = hb + (size_t)ch1 * kChanStride + ((it & 63) << 6) + (jt & 63);
    lh1 = __builtin_nontemporal_load(d1);
    hl1 = __builtin_nontemporal_load(d1 + 4 * kChanStride);
    hh1 = __builtin_nontemporal_load(d1 + 8 * kChanStride);
  }

  // ---- Level-0 details for the 2x2 level-0 positions (i2,j2) = (2it+r, 2jt+u)
  //      Adjacent u share a float2 (x0 even => 8B aligned); bands 16 ch apart.
  const int pw0 = jt >> 5;               // (2*jt) >> 6
  const int x0  = (jt << 1) & 63;
  v2f lh0[2], hl0[2], hh0[2];
#pragma unroll
  for (int r = 0; r < 2; ++r) {
    const int i2  = (it << 1) | r;
    const int ch0 = 3 + c * 48 + ((i2 >> 6) << 2) + pw0;
    const float* d0 = hb + (size_t)ch0 * kChanStride + ((i2 & 63) << 6) + x0;
    lh0[r] = __builtin_nontemporal_load((const v2f*)d0);
    hl0[r] = __builtin_nontemporal_load((const v2f*)(d0 + 16 * kChanStride));
    hh0[r] = __builtin_nontemporal_load((const v2f*)(d0 + 32 * kChanStride));
  }

  // ---- Final synthesis + 4 aligned float4 row stores (Y = 4*it + 2r + s)
  const size_t nc = (size_t)n * 3 + c;
  float* ob = out + ((nc * 512 + (size_t)(it << 2)) << 9) + (jt << 2);
#pragma unroll
  for (int r = 0; r < 2; ++r) {
    const float ar = r ? -1.f : 1.f;
    const float e = v1 + ar * lh1;
    const float f = hl1 + ar * hh1;
    const float v2u0 = 0.5f * (e + f);   // ll2 at (2it+r, 2jt)
    const float v2u1 = 0.5f * (e - f);   // ll2 at (2it+r, 2jt+1)
#pragma unroll
    for (int s = 0; s < 2; ++s) {
      const float S = s ? -1.f : 1.f;
      v4f w;
      const float p0 = v2u0 + S * lh0[r].x;
      const float m0 = hl0[r].x + S * hh0[r].x;
      w.x = 0.5f * (p0 + m0);
      w.y = 0.5f * (p0 - m0);
      const float p1 = v2u1 + S * lh0[r].y;
      const float m1 = hl0[r].y + S * hh0[r].y;
      w.z = 0.5f * (p1 + m1);
      w.w = 0.5f * (p1 - m1);
      __builtin_nontemporal_store(w, (v4f*)(ob + (size_t)(((r << 1) | s) << 9)));
    }
  }
}

extern "C" void kernel_launch(void* const* d_in, const int* in_sizes, int n_in,
                              void* d_out, int out_size, void* d_ws, size_t ws_size,
                              hipStream_t stream) {
  (void)n_in; (void)out_size; (void)d_ws; (void)ws_size;
  const float* h = (const float*)d_in[0];
  float* out = (float*)d_out;
  const int n = in_sizes[0] / kInNStride;   // batch (16 for the reference shapes)
  dim3 block(128, 2, 1);                    // 256 threads = 8 wave32
  dim3 grid(3, 64, n);                      // c, i0 row, batch  (no div-by-3)
  haar_idwt3_fused<<<grid, block, 0, stream>>>(h, out);
}